// NLSPNModel_Adapt_17008070492331
// MI455X (gfx1250) — compile-verified
//
#include <hip/hip_runtime.h>
#include <math.h>

#define BN 4
#define HN 352
#define WN 1216
#define HW (HN * WN)
#define NPIX (BN * HW)            // 1,712,128
#define GCH 8
#define KTOT 72                   // 8 ch * 3 * 3
#define TILEW 32
#define NTX (WN / TILEW)          // 38
#define NTILES (BN * HN * NTX)    // 53,504 wave-tiles
#define WPB 8                     // waves per block
#define PREP_BLOCKS (NTILES / WPB)  // 6688 (exact)
#define PROP_BLOCKS (NPIX / 256)    // 6688 (exact)

#define GT_ROWSTRIDE 36           // 34 cols + 2 pad
#define GT_CHSTRIDE (3 * GT_ROWSTRIDE)   // 108
#define GT_SIZE (GCH * GT_CHSTRIDE)      // 864 floats per wave tile
#define CSTRIDE 33                // C buffer [32 pix][33] per wave
#define CSLICE (32 * CSTRIDE)     // 1056 floats per wave
#define WSTRIDE 76                // W_oa LDS row stride (conflict-free mod 64)

typedef __attribute__((ext_vector_type(8))) float v8f;
typedef __attribute__((ext_vector_type(2))) float v2f;

// LDS address (within a wave's guidance tile) for im2col index K = ci*9 + kh*3 + kw,
// with the kw column shift folded in. Constant-folded after full unroll.
__host__ __device__ constexpr int kaddr(int K) {
  return (K / 9) * GT_CHSTRIDE + ((K % 9) / 3) * GT_ROWSTRIDE + ((K % 9) % 3);
}

// ---------- helpers ----------

__device__ __forceinline__ float ld_guid(const float* __restrict__ g,
                                         int b, int i, int ys, int xs) {
  if (ys < 0 || ys >= HN || xs < 0 || xs >= WN) return 0.0f;
  return g[((b * GCH + i) * HN + ys) * WN + xs];
}

__device__ __forceinline__ float samp0(const float* __restrict__ fb, int yi, int xi) {
  if (yi < 0 || yi >= HN || xi < 0 || xi >= WN) return 0.0f;
  return fb[yi * WN + xi];
}

__device__ __forceinline__ float bilinear(const float* __restrict__ fb, float ys, float xs) {
  float y0f = floorf(ys), x0f = floorf(xs);
  float wy = ys - y0f, wx = xs - x0f;
  int y0 = (int)y0f, x0 = (int)x0f;
  float v00 = samp0(fb, y0,     x0);
  float v01 = samp0(fb, y0,     x0 + 1);
  float v10 = samp0(fb, y0 + 1, x0);
  float v11 = samp0(fb, y0 + 1, x0 + 1);
  return (1.0f - wy) * ((1.0f - wx) * v00 + wx * v01) +
         wy          * ((1.0f - wx) * v10 + wx * v11);
}

// ---------- kernel 1: fused conv(8->24, 3x3) via WMMA + offset/aff prep ----------
// Wave handles 32 consecutive x of one (b, y) row.
// A0 = pixels 0..15, A1 = pixels 16..31 ; B0 = out-ch 0..15, B1 = out-ch 16..31 (24..31 zero).

__global__ void nlspn_prep(const float* __restrict__ guidance,
                           const float* __restrict__ confidence,
                           const float* __restrict__ feat_init,
                           const float* __restrict__ W_oa,
                           const float* __restrict__ b_oa,
                           const float* __restrict__ aff_scale,
                           float* __restrict__ out_off,   // (B,18,H,W)
                           float* __restrict__ out_aff,   // (B,9,H,W)
                           float* __restrict__ feat0) {   // (B,H,W)
  // cbuf: per-wave slice reused as (a) guidance halo tile, then (b) C transpose buffer.
  __shared__ float cbuf[WPB * CSLICE];      // 8448 floats
  __shared__ float wlds[32 * WSTRIDE];      // 2432 floats, zero-padded weights

  const int wave  = threadIdx.x >> 5;
  const int lane  = threadIdx.x & 31;
  const int wtile = blockIdx.x * WPB + wave;   // grid is exact: wtile < NTILES

  const int tx    = wtile % NTX;
  const int rowid = wtile / NTX;
  const int y     = rowid % HN;
  const int b     = rowid / HN;
  const int xbase = tx * TILEW;

  const int m  = lane & 15;       // M (pixel) / N (channel) within half-tile
  const int lg = lane >> 4;       // lane group selects K pair

  float* slice = cbuf + wave * CSLICE;

  // ---- stage guidance halo tile (zero-padded) into this wave's LDS slice ----
  for (int t = lane; t < GT_SIZE; t += 32) {
    const int ci  = t / GT_CHSTRIDE;
    const int r   = t - ci * GT_CHSTRIDE;
    const int row = r / GT_ROWSTRIDE;
    const int col = r - row * GT_ROWSTRIDE;
    float v = 0.0f;
    if (col < 34) v = ld_guid(guidance, b, ci, y + row - 1, xbase + col - 1);
    slice[t] = v;
  }
  // ---- stage weights (zero-padded to 32 out-ch x 76) once per block ----
  for (int t = threadIdx.x; t < 32 * WSTRIDE; t += 256) {
    const int n = t / WSTRIDE;
    const int K = t - n * WSTRIDE;
    wlds[t] = (n < 24 && K < KTOT) ? W_oa[n * KTOT + K] : 0.0f;
  }
  __syncthreads();

  // ---- WMMA main loop: fully unrolled, LDS-only operand feed ----
  v8f acc00 = {}, acc01 = {}, acc10 = {}, acc11 = {};
  #pragma unroll
  for (int k0 = 0; k0 < KTOT; k0 += 4) {
    v2f a0, a1, bb0, bb1;
    #pragma unroll
    for (int v = 0; v < 2; ++v) {
      const int Klo = k0 + v;          // compile-time
      const int Khi = k0 + v + 2;      // compile-time
      const int ka  = lg ? kaddr(Khi) : kaddr(Klo);   // one v_cndmask
      const int Kb  = lg ? Khi : Klo;                 // one v_cndmask
      a0[v]  = slice[ka + m];          // pixel p = m
      a1[v]  = slice[ka + 16 + m];     // pixel p = 16 + m
      bb0[v] = wlds[m * WSTRIDE + Kb];
      bb1[v] = wlds[(16 + m) * WSTRIDE + Kb];
    }
    acc00 = __builtin_amdgcn_wmma_f32_16x16x4_f32(false, a0, false, bb0, (short)0, acc00, false, false);
    acc01 = __builtin_amdgcn_wmma_f32_16x16x4_f32(false, a0, false, bb1, (short)0, acc01, false, false);
    acc10 = __builtin_amdgcn_wmma_f32_16x16x4_f32(false, a1, false, bb0, (short)0, acc10, false, false);
    acc11 = __builtin_amdgcn_wmma_f32_16x16x4_f32(false, a1, false, bb1, (short)0, acc11, false, false);
  }
  __syncthreads();   // all gtile reads done before overwriting slice with C

  // ---- scatter C layout (lane = N channel, VGPR = M pixel) -> [pixel][channel] ----
  #pragma unroll
  for (int v = 0; v < 8; ++v) {
    const int pA = v + 8 * lg;
    slice[pA * CSTRIDE        + m]      = acc00[v];
    slice[pA * CSTRIDE        + 16 + m] = acc01[v];
    slice[(16 + pA) * CSTRIDE + m]      = acc10[v];
    slice[(16 + pA) * CSTRIDE + 16 + m] = acc11[v];
  }
  __syncthreads();

  // ---- per-pixel epilogue: lane owns pixel x = xbase + lane ----
  const int x = xbase + lane;
  float oa[24];
  #pragma unroll
  for (int n = 0; n < 24; ++n) oa[n] = slice[lane * CSTRIDE + n] + b_oa[n];

  // off_y[k] = oa[2k], off_x[k] = oa[2k+1]; zero inserted at j = 4
  float offy[9], offx[9];
  #pragma unroll
  for (int j = 0; j < 9; ++j) {
    if (j == 4) { offy[j] = 0.0f; offx[j] = 0.0f; }
    else {
      const int k = (j < 4) ? j : j - 1;
      offy[j] = oa[2 * k];
      offx[j] = oa[2 * k + 1];
    }
  }

  const float scale = aff_scale[0] + 1e-8f;
  const float* confb = confidence + b * HW;
  float a8[8];
  float s = 0.0f;
  #pragma unroll
  for (int k = 0; k < 8; ++k) {
    const int j = (k < 4) ? k : k + 1;
    float a = tanhf(oa[16 + k]) / scale;
    a *= bilinear(confb, (float)y + offy[j], (float)x + offx[j]);
    a8[k] = a;
    s += fabsf(a);
  }
  const float denom = fmaxf(s + 1e-4f, 1.0f);
  float asum = 0.0f;
  #pragma unroll
  for (int k = 0; k < 8; ++k) { a8[k] /= denom; asum += a8[k]; }

  const int pix = y * WN + x;
  #pragma unroll
  for (int j = 0; j < 9; ++j) {
    out_off[((b * 18 + 2 * j)     * HN + y) * WN + x] = offy[j];
    out_off[((b * 18 + 2 * j + 1) * HN + y) * WN + x] = offx[j];
    const float av = (j == 4) ? (1.0f - asum) : a8[(j < 4) ? j : j - 1];
    out_aff[((b * 9 + j) * HN + y) * WN + x] = av;
  }
  feat0[b * HW + pix] = feat_init[b * HW + pix];
}

// ---------- kernel 2: one propagation step (9-tap bilinear gather) ----------

__global__ void nlspn_prop(const float* __restrict__ feat_in,
                           float* __restrict__ feat_out,
                           const float* __restrict__ off,   // (B,18,H,W)
                           const float* __restrict__ aff) { // (B,9,H,W)
  const int idx = blockIdx.x * 256 + threadIdx.x;
  if (idx >= NPIX) return;
  const int x = idx % WN;
  const int t = idx / WN;
  const int y = t % HN;
  const int b = t / HN;
  const float* fb = feat_in + b * HW;

  // warm the rows we are about to gather from (global_prefetch_b8 path)
  __builtin_prefetch(fb + (y > 0 ? y - 1 : 0) * WN + x, 0, 1);
  __builtin_prefetch(fb + (y < HN - 1 ? y + 1 : y) * WN + x, 0, 1);

  float acc = 0.0f;
  #pragma unroll
  for (int j = 0; j < 9; ++j) {
    const float a  = aff[((b * 9 + j) * HN + y) * WN + x];
    const float oy = off[((b * 18 + 2 * j)     * HN + y) * WN + x];
    const float ox = off[((b * 18 + 2 * j + 1) * HN + y) * WN + x];
    const float ys = (float)y + (float)(j / 3 - 1) + oy;
    const float xs = (float)x + (float)(j % 3 - 1) + ox;
    acc += a * bilinear(fb, ys, xs);
  }
  feat_out[idx] = acc;
}

// ---------- host launch ----------

extern "C" void kernel_launch(void* const* d_in, const int* in_sizes, int n_in,
                              void* d_out, int out_size, void* d_ws, size_t ws_size,
                              hipStream_t stream) {
  const float* guidance   = (const float*)d_in[0];
  const float* confidence = (const float*)d_in[1];
  const float* feat_init  = (const float*)d_in[2];
  const float* W_oa       = (const float*)d_in[3];
  const float* b_oa       = (const float*)d_in[4];
  const float* aff_scale  = (const float*)d_in[5];

  float* out      = (float*)d_out;
  float* out_feat = out;                 // (B,1,H,W)
  float* out_off  = out + NPIX;          // (B,18,H,W)
  float* out_aff  = out + 19 * NPIX;     // (B,9,H,W)

  float* ws0 = (float*)d_ws;             // feat ping
  float* ws1 = ws0 + NPIX;               // feat pong

  nlspn_prep<<<PREP_BLOCKS, 256, 0, stream>>>(guidance, confidence, feat_init,
                                              W_oa, b_oa, aff_scale,
                                              out_off, out_aff, ws0);

  for (int it = 0; it < 18; ++it) {
    const float* src = (it % 2 == 0) ? ws0 : ws1;
    float* dst = (it == 17) ? out_feat : ((it % 2 == 0) ? ws1 : ws0);
    nlspn_prop<<<PROP_BLOCKS, 256, 0, stream>>>(src, dst, out_off, out_aff);
  }
}